// PointNetSetAbstractionMsg_70239895159140
// MI455X (gfx1250) — compile-verified
//
#include <hip/hip_runtime.h>
#include <hip/hip_bf16.h>

typedef float v2f __attribute__((ext_vector_type(2)));
typedef float v8f __attribute__((ext_vector_type(8)));

#define BB 8
#define NN 4096
#define DD 64
#define SS 1024          // NPOINT
#define FC 68            // 3 + 64 + 1 pad
#define EPSBN 1e-5f

// deterministic in-wave argmax (max value, lowest index on ties)
__device__ __forceinline__ void wave_argmax(float& v, int& i) {
#pragma unroll
  for (int off = 16; off; off >>= 1) {
    float ov = __shfl_xor(v, off);
    int oi = __shfl_xor(i, off);
    if (ov > v || (ov == v && oi < i)) { v = ov; i = oi; }
  }
}

// ---------------- FPS ----------------
// one block/batch, 1024 threads, 4 pts/thread in regs; 2 barriers per step
__global__ __launch_bounds__(1024) void fps_kernel(
    const float* __restrict__ xyz,   // [B,3,N]
    float* __restrict__ new_xyz_t,   // ws [B,S,3]
    float* __restrict__ out_newxyz)  // d_out [B,3,S]
{
  __shared__ float px[NN], py[NN], pz[NN];
  __shared__ float wval[32];
  __shared__ int   widx[32];
  __shared__ int   sfar;
  const int b = blockIdx.x, tid = threadIdx.x;
  const int wave = tid >> 5, lane = tid & 31;
  const float* xb = xyz + (size_t)b * 3 * NN;
  for (int j = tid; j < NN; j += 1024) {
    px[j] = xb[j]; py[j] = xb[NN + j]; pz[j] = xb[2 * NN + j];
  }
  if (tid == 0) sfar = 0;
  __syncthreads();
  const int base = tid * 4;
  float d0 = 1e10f, d1 = 1e10f, d2 = 1e10f, d3 = 1e10f;
  for (int i = 0; i < SS; ++i) {
    const int far = sfar;
    const float cx = px[far], cy = py[far], cz = pz[far];
    if (tid == 0) {
      float* nx = new_xyz_t + ((size_t)b * SS + i) * 3;
      nx[0] = cx; nx[1] = cy; nx[2] = cz;
      out_newxyz[(size_t)b * 3 * SS + 0 * SS + i] = cx;
      out_newxyz[(size_t)b * 3 * SS + 1 * SS + i] = cy;
      out_newxyz[(size_t)b * 3 * SS + 2 * SS + i] = cz;
    }
    float e, t;
    e = px[base+0]-cx; t = e*e; e = py[base+0]-cy; t += e*e; e = pz[base+0]-cz; t += e*e; d0 = fminf(d0, t);
    e = px[base+1]-cx; t = e*e; e = py[base+1]-cy; t += e*e; e = pz[base+1]-cz; t += e*e; d1 = fminf(d1, t);
    e = px[base+2]-cx; t = e*e; e = py[base+2]-cy; t += e*e; e = pz[base+2]-cz; t += e*e; d2 = fminf(d2, t);
    e = px[base+3]-cx; t = e*e; e = py[base+3]-cy; t += e*e; e = pz[base+3]-cz; t += e*e; d3 = fminf(d3, t);
    float bv = d0; int bi = base;
    if (d1 > bv) { bv = d1; bi = base + 1; }
    if (d2 > bv) { bv = d2; bi = base + 2; }
    if (d3 > bv) { bv = d3; bi = base + 3; }
    wave_argmax(bv, bi);
    if (lane == 0) { wval[wave] = bv; widx[wave] = bi; }
    __syncthreads();
    if (wave == 0) {
      float v = wval[lane]; int ix = widx[lane];
      wave_argmax(v, ix);
      if (lane == 0) sfar = ix;
    }
    __syncthreads();
  }
}

// ---------------- Ball query ----------------
__global__ void ballquery_kernel(const float* __restrict__ xyz,
                                 const float* __restrict__ new_xyz_t,
                                 int* __restrict__ gidx, float r2, int ns)
{
  int g = blockIdx.x * blockDim.x + threadIdx.x;
  if (g >= BB * SS) return;
  int b = g / SS;
  const float* c = new_xyz_t + (size_t)g * 3;
  float cx = c[0], cy = c[1], cz = c[2];
  const float* xb = xyz + (size_t)b * 3 * NN;
  int cnt = 0, first = 0;
  long base = (long)g * ns;
  for (int j = 0; j < NN; ++j) {
    float dx = xb[j] - cx, dy = xb[NN + j] - cy, dz = xb[2 * NN + j] - cz;
    if (dx * dx + dy * dy + dz * dz <= r2) {
      if (cnt == 0) first = j;
      gidx[base + cnt] = j;
      if (++cnt == ns) break;
    }
  }
  for (int k = cnt; k < ns; ++k) gidx[base + k] = first;
}

// ---------------- One-time transpose: feat_t[B,N,68] = {xyz_t | points_t | 0} ---
__global__ void build_feat(const float* __restrict__ xyz,
                           const float* __restrict__ points,
                           float* __restrict__ feat)
{
  long idx = (long)blockIdx.x * blockDim.x + threadIdx.x;
  const long total = (long)BB * NN * FC;
  if (idx >= total) return;
  int c = (int)(idx % FC);
  long bn = idx / FC;
  int b = (int)(bn / NN), n = (int)(bn % NN);
  float v;
  if (c < 3)            v = xyz[(size_t)b * 3 * NN + (size_t)c * NN + n];
  else if (c < 3 + DD)  v = points[(size_t)b * DD * NN + (size_t)(c - 3) * NN + n];
  else                  v = 0.f;
  feat[idx] = v;
}

// ---------------- Gather (contiguous rows from feat_t) ----------------
__global__ void gather_kernel(const float* __restrict__ feat,
                              const float* __restrict__ new_xyz_t,
                              const int* __restrict__ gidx,
                              float* __restrict__ X, int ns, long total)
{
  long idx = (long)blockIdx.x * blockDim.x + threadIdx.x;
  if (idx >= total) return;
  long m = idx / FC;
  int c = (int)(idx % FC);
  int b = (int)(m / ((long)SS * ns));
  long rem = m % ((long)SS * ns);
  int s = (int)(rem / ns);
  int p = gidx[m];
  float v = feat[((size_t)b * NN + p) * FC + c];
  if (c < 3) v -= new_xyz_t[((size_t)b * SS + s) * 3 + c];
  X[idx] = v;
}

// ---------------- Fused GEMM: prevBN+ReLU on A-load, WMMA f32, stats epilogue ---
// Block: 8 waves = 128 rows. Whole W in LDS. Each wave: 16-row A cached in regs,
// loops all N-tiles (A reused). Deterministic per-block channel partial sums.
template <int KP, int COUT>
__global__ __launch_bounds__(256) void gemm_fused(
    const float* __restrict__ X,        // [M, KP] raw previous-layer output
    const float* __restrict__ W,        // [COUT, cin] unpadded
    int cin,
    float* __restrict__ Y,              // [M, COUT] raw output
    const float* __restrict__ statsPrev,// null for layer 0 (no BN/ReLU on input)
    const float* __restrict__ gammaPrev,
    const float* __restrict__ betaPrev,
    long M,
    float* __restrict__ partial)        // [gridDim.x, 256] per-channel {sum, sumsq}
{
  __shared__ float lw[COUT * KP];  // staged weights, rows padded to KP
  __shared__ float scsh[2 * 96];   // prev scale / shift
  __shared__ float part[8][256];   // wave-private per-channel {sum, sumsq}
  const int tid = threadIdx.x;
  const bool hasPrev = (statsPrev != nullptr);

  if (hasPrev) {
    for (int c = tid; c < cin; c += 256) {
      float inv = 1.f / (float)M;
      float mean = statsPrev[c] * inv;
      float var = statsPrev[128 + c] * inv - mean * mean;
      float s = gammaPrev[c] * rsqrtf(var + EPSBN);
      scsh[c] = s;
      scsh[96 + c] = betaPrev[c] - mean * s;
    }
  }
  for (int i = tid; i < COUT * KP; i += 256) {
    int rr = i / KP, kk = i % KP;
    lw[i] = (kk < cin) ? W[(size_t)rr * cin + kk] : 0.f;
  }
  __syncthreads();

  const int wave = tid >> 5, lane = tid & 31;
  const int row0 = (blockIdx.x * 8 + wave) * 16;
  const int mrow = row0 + (lane & 15);
  const int koff = (lane >> 4) * 2;   // lanes 0-15 -> K{0,1}; 16-31 -> K{2,3}

  v2f a_reg[KP / 4];
  const float* xrow = X + (size_t)mrow * KP + koff;
#pragma unroll
  for (int ks = 0; ks < KP / 4; ++ks) {
    v2f a = *(const v2f*)(xrow + ks * 4);
    if (hasPrev) {
      int c0 = ks * 4 + koff;
      float v0 = a.x * scsh[c0] + scsh[96 + c0];
      float v1 = a.y * scsh[c0 + 1] + scsh[96 + c0 + 1];
      a.x = v0 > 0.f ? v0 : 0.f;
      a.y = v1 > 0.f ? v1 : 0.f;
    }
    a_reg[ks] = a;
  }

  constexpr int NT = COUT >> 4;
  float* yw = Y + (size_t)(row0 + (lane >> 4) * 8) * COUT + (lane & 15);
#pragma unroll
  for (int nt = 0; nt < NT; ++nt) {
    const float* wrow = lw + (size_t)(nt * 16 + (lane & 15)) * KP + koff;
    v8f acc = {0.f, 0.f, 0.f, 0.f, 0.f, 0.f, 0.f, 0.f};
#pragma unroll
    for (int ks = 0; ks < KP / 4; ++ks) {
      v2f bfrag = *(const v2f*)(wrow + ks * 4);
      acc = __builtin_amdgcn_wmma_f32_16x16x4_f32(
          false, a_reg[ks], false, bfrag, (short)0, acc, false, false);
    }
    float s = 0.f, sq = 0.f;
#pragma unroll
    for (int r = 0; r < 8; ++r) {
      float v = acc[r];
      yw[(size_t)r * COUT + nt * 16] = v;
      s += v;
      sq += v * v;
    }
    s  += __shfl_xor(s, 16);
    sq += __shfl_xor(sq, 16);
    if (lane < 16) {
      int c = nt * 16 + lane;
      part[wave][c] = s;          // each column written exactly once per wave
      part[wave][128 + c] = sq;
    }
  }
  __syncthreads();
  for (int c = tid; c < COUT; c += 256) {
    float s = 0.f, sq = 0.f;
#pragma unroll
    for (int w = 0; w < 8; ++w) { s += part[w][c]; sq += part[w][128 + c]; }
    partial[(size_t)blockIdx.x * 256 + c] = s;
    partial[(size_t)blockIdx.x * 256 + 128 + c] = sq;
  }
}

// ---------------- Deterministic two-stage stats reduction ----------------
__global__ __launch_bounds__(256) void bn_reduce1(const float* __restrict__ partial,
                                                  int nblocks, int cout,
                                                  float* __restrict__ partial2)
{
  int c = threadIdx.x;
  int i0 = blockIdx.x * 256;
  int i1 = min(i0 + 256, nblocks);
  if ((c & 127) < cout) {
    float s = 0.f;
    for (int i = i0; i < i1; ++i) s += partial[(size_t)i * 256 + c];
    partial2[(size_t)blockIdx.x * 256 + c] = s;
  }
}

__global__ __launch_bounds__(256) void bn_reduce2(const float* __restrict__ partial2,
                                                  int nchunks, int cout,
                                                  float* __restrict__ stats)
{
  int c = threadIdx.x;
  if ((c & 127) < cout) {
    float s = 0.f;
    for (int i = 0; i < nchunks; ++i) s += partial2[(size_t)i * 256 + c];
    stats[c] = s;
  }
}

// ---------------- Max over ns with fused final BN+ReLU ----------------
__global__ __launch_bounds__(256) void maxpool_fused(
    const float* __restrict__ Y, int C, int ns,
    const float* __restrict__ stats,
    const float* __restrict__ gamma, const float* __restrict__ beta, long M,
    float* __restrict__ out, int obase, long total)
{
  __shared__ float sc[128], sh[128];
  int tid = threadIdx.x;
  if (tid < C) {
    float inv = 1.f / (float)M;
    float mean = stats[tid] * inv;
    float var = stats[128 + tid] * inv - mean * mean;
    float s = gamma[tid] * rsqrtf(var + EPSBN);
    sc[tid] = s;
    sh[tid] = beta[tid] - mean * s;
  }
  __syncthreads();
  long g = (long)blockIdx.x * blockDim.x + tid;
  if (g >= total) return;
  int c = (int)(g % C);
  long bs = g / C;
  int b = (int)(bs / SS), s = (int)(bs % SS);
  size_t base = (size_t)bs * ns * C + c;
  float scl = sc[c], shf = sh[c];
  float m = -1e30f;
  for (int j = 0; j < ns; ++j) {
    float v = Y[base + (size_t)j * C] * scl + shf;
    v = v > 0.f ? v : 0.f;
    m = fmaxf(m, v);
  }
  const size_t NPOFF = (size_t)BB * 3 * SS;
  out[NPOFF + ((size_t)b * 320 + obase + c) * SS + s] = m;
}

// ---------------- Host launch ----------------
static const int   H_NS[3]      = {16, 32, 64};
static const float H_R2[3]      = {0.01f, 0.04f, 0.16f};
static const int   H_COUT[3][3] = {{32, 32, 64}, {64, 64, 128}, {64, 96, 128}};
static const int   H_OBASE[3]   = {0, 64, 192};

static void launch_gemm(int KP, int cout, dim3 grid, hipStream_t stream,
                        const float* X, const float* W, int cin,
                        float* Y, const float* sp, const float* gp, const float* bp,
                        long M, float* partial) {
#define GEMM_CASE(kp, co)                                                        \
  if (KP == kp && cout == co) {                                                  \
    gemm_fused<kp, co><<<grid, 256, 0, stream>>>(X, W, cin, Y, sp, gp, bp, M,    \
                                                 partial);                       \
    return;                                                                      \
  }
  GEMM_CASE(68, 32)
  GEMM_CASE(68, 64)
  GEMM_CASE(32, 32)
  GEMM_CASE(32, 64)
  GEMM_CASE(64, 64)
  GEMM_CASE(64, 96)
  GEMM_CASE(64, 128)
  GEMM_CASE(96, 128)
#undef GEMM_CASE
}

extern "C" void kernel_launch(void* const* d_in, const int* in_sizes, int n_in,
                              void* d_out, int out_size, void* d_ws, size_t ws_size,
                              hipStream_t stream) {
  const float* xyz = nullptr;
  const float* points = nullptr;
  int pl[48]; int np = 0;
  for (int i = 0; i < n_in; ++i) {
    if (in_sizes[i] == BB * 3 * NN && !xyz) { xyz = (const float*)d_in[i]; }
    else if (in_sizes[i] == BB * DD * NN && !points) { points = (const float*)d_in[i]; }
    else if (np < 48) { pl[np++] = i; }
  }
  const float *Wp[3][3], *Gp[3][3], *Betap[3][3];
  int li = 0;
  for (int br = 0; br < 3; ++br) {
    int cin = DD + 3;
    for (int l = 0; l < 3; ++l) {
      int cout = H_COUT[br][l];
      int idxs[4] = {pl[li], pl[li + 1], pl[li + 2], pl[li + 3]};
      li += 4;
      int wslot = 0;
      for (int t = 0; t < 4; ++t) if (in_sizes[idxs[t]] == cout * cin) wslot = t;
      Wp[br][l] = (const float*)d_in[idxs[wslot]];
      int vs[3], vn = 0;
      for (int t = 0; t < 4; ++t) if (t != wslot) vs[vn++] = t;
      // alphabetical leaf order within layer dict: b, beta, gamma
      Betap[br][l] = (const float*)d_in[idxs[vs[1]]];
      Gp[br][l]   = (const float*)d_in[idxs[vs[2]]];
      cin = cout;
    }
  }

  char* ws = (char*)d_ws;
  float* new_xyz_t = (float*)(ws + (64ull << 10));
  int*   gidx      = (int*)(ws + (256ull << 10));
  float* stats_buf = (float*)(ws + (5ull << 20) / 2);              // 3 * 256 floats
  float* partial2  = (float*)(ws + (5ull << 20) / 2 + (256ull << 10));
  float* partial   = (float*)(ws + (3ull << 20));                  // up to 4 MiB
  float* feat      = (float*)(ws + (8ull << 20));                  // ~8.9 MiB
  float* bufA      = (float*)(ws + (20ull << 20));                 // up to 192 MiB
  float* bufB      = (float*)(ws + (20ull << 20) + (208ull << 20));// up to 256 MiB
  float* out = (float*)d_out;

  fps_kernel<<<BB, 1024, 0, stream>>>(xyz, new_xyz_t, out);
  {
    long ft = (long)BB * NN * FC;
    build_feat<<<(unsigned)((ft + 255) / 256), 256, 0, stream>>>(xyz, points, feat);
  }

  for (int br = 0; br < 3; ++br) {
    const int ns = H_NS[br];
    const long M = (long)BB * SS * ns;
    ballquery_kernel<<<(BB * SS + 255) / 256, 256, 0, stream>>>(xyz, new_xyz_t, gidx, H_R2[br], ns);

    long gtot = M * FC;
    gather_kernel<<<(unsigned)((gtot + 255) / 256), 256, 0, stream>>>(
        feat, new_xyz_t, gidx, bufA, ns, gtot);

    int cin = DD + 3;
    const float* Xc = bufA; int KP = FC;
    for (int l = 0; l < 3; ++l) {
      int cout = H_COUT[br][l];
      float* Yc = (l & 1) ? bufA : bufB;   // l0->B, l1->A, l2->B
      int nblocks = (int)(M / 128);
      const float* sp = (l == 0) ? nullptr : stats_buf + (l - 1) * 256;
      const float* gp = (l == 0) ? nullptr : Gp[br][l - 1];
      const float* bp = (l == 0) ? nullptr : Betap[br][l - 1];
      launch_gemm(KP, cout, dim3((unsigned)nblocks), stream, Xc, Wp[br][l], cin,
                  Yc, sp, gp, bp, M, partial);
      int nchunks = (nblocks + 255) / 256;
      bn_reduce1<<<(unsigned)nchunks, 256, 0, stream>>>(partial, nblocks, cout, partial2);
      bn_reduce2<<<1, 256, 0, stream>>>(partial2, nchunks, cout, stats_buf + l * 256);
      Xc = Yc; KP = cout; cin = cout;
    }
    int cl = H_COUT[br][2];
    long mtot = (long)BB * SS * cl;
    maxpool_fused<<<(unsigned)((mtot + 255) / 256), 256, 0, stream>>>(
        bufB, cl, ns, stats_buf + 2 * 256, Gp[br][2], Betap[br][2], M,
        out, H_OBASE[br], mtot);
  }
}